// SpikeSearchLoss_32968168964665
// MI455X (gfx1250) — compile-verified
//
#include <hip/hip_runtime.h>

typedef __attribute__((ext_vector_type(2))) float v2f;
typedef __attribute__((ext_vector_type(8))) float v8f;

#define NPTS 1024
#define NTILE 64                          // 1024 / 16
#define NTILEPAIRS (NTILE * (NTILE + 1) / 2)  // 2080, divisible by 8 waves
#define XSTRIDE 20
#define XCOLS 20480

__global__ void sl_init_kernel(unsigned int* ws) {
    if (threadIdx.x == 0) ws[0] = 0x7F800000u; // +inf bit pattern
}

__global__ __launch_bounds__(256)
void sl_min_pairdist_kernel(const float* __restrict__ x, unsigned int* __restrict__ ws) {
    __shared__ float lpx[NPTS];
    __shared__ float lpy[NPTS];
    __shared__ float lsq[NPTS];
    __shared__ float red[256];

    const int b   = blockIdx.x;
    const int tid = threadIdx.x;
    const float* __restrict__ xb = x + (size_t)b * XCOLS;

    // Stage strided points into LDS; precompute squared norms.
    for (int k = tid; k < NPTS; k += 256) {
        float px = xb[k * XSTRIDE];
        float py = xb[k * XSTRIDE + 1];
        lpx[k] = px;
        lpy[k] = py;
        lsq[k] = px * px + py * py;
    }
    __syncthreads();

    const int  lane = tid & 31;   // wave32
    const int  wave = tid >> 5;   // 8 waves per block
    const int  col  = lane & 15;
    const bool lo   = lane < 16;  // lanes 16..31 hold the zero K=2,3 slices

    float lmin = __builtin_inff();

    // Each wave handles tile pairs t = wave, wave+8, ... (2080 total -> 260 each,
    // perfectly uniform: EXEC stays all-ones around the WMMA).
    for (int t = wave; t < NTILEPAIRS; t += 8) {
        // Map linear t -> (I, J) with I <= J over the 64x64 tile grid.
        int I = 0, base = 0;
        while (t >= base + (NTILE - I)) { base += NTILE - I; ++I; }
        const int J = I + (t - base);

        // A: 16x4 f32 (rows = tile-I points, K = {x, y, 0, 0})
        // B: 4x16 f32 (cols = tile-J points); ISA layout: lanes 0-15 = K 0..1
        // across VGPR0/1, lanes 16-31 = K 2..3 (zeros).
        v2f av, bv;
        {
            const int ri = I * 16 + col;
            const int ci = J * 16 + col;
            av.x = lo ? lpx[ri] : 0.0f;
            av.y = lo ? lpy[ri] : 0.0f;
            bv.x = lo ? lpx[ci] : 0.0f;
            bv.y = lo ? lpy[ci] : 0.0f;
        }

        v8f c = {};
        // D = A x B + 0  -> 16x16 Gram tile of dot(pos_i, pos_j)
        c = __builtin_amdgcn_wmma_f32_16x16x4_f32(
                /*neg_a=*/false, av, /*neg_b=*/false, bv,
                /*c_mod=*/(short)0, c, /*reuse_a=*/false, /*reuse_b=*/false);

        const int   j    = J * 16 + col;
        const float sqj  = lsq[j];
        const int rbase  = I * 16 + (lo ? 0 : 8);  // C/D layout: VGPR v -> M = v (+8 hi lanes)
#pragma unroll
        for (int v = 0; v < 8; ++v) {
            const int   i  = rbase + v;
            const float d2 = lsq[i] + sqj - 2.0f * c[v];
            const bool  ok = (i < j) && (d2 > 0.0f);
            lmin = (ok && d2 < lmin) ? d2 : lmin;
        }
    }

    // Block min-reduction.
    red[tid] = lmin;
    __syncthreads();
    for (int s = 128; s > 0; s >>= 1) {
        if (tid < s) red[tid] = fminf(red[tid], red[tid + s]);
        __syncthreads();
    }
    if (tid == 0) {
        // Non-negative IEEE floats order identically to their uint bit patterns.
        atomicMin(ws, __float_as_uint(red[0]));
    }
}

__global__ void sl_finalize_kernel(const float* __restrict__ Y,
                                   const unsigned int* __restrict__ ws,
                                   float* __restrict__ out) {
    __shared__ float s[64];
    const int tid = threadIdx.x;
    s[tid] = Y[tid];
    __syncthreads();
    if (tid == 0) {
        float sum = 0.0f;
        for (int i = 0; i < 64; ++i) sum += s[i];
        const float meanY = sum * (1.0f / 64.0f);
        const float mind2 = __uint_as_float(ws[0]);
        out[0] = (1.0f - meanY) + 5e-5f / sqrtf(mind2);
    }
}

extern "C" void kernel_launch(void* const* d_in, const int* in_sizes, int n_in,
                              void* d_out, int out_size, void* d_ws, size_t ws_size,
                              hipStream_t stream) {
    const float* x  = (const float*)d_in[0];   // (64, 20480) fp32
    const float* Y  = (const float*)d_in[1];   // (64, 1) fp32
    float* out      = (float*)d_out;           // scalar fp32
    unsigned int* ws = (unsigned int*)d_ws;    // [0] = running min (uint bits)

    sl_init_kernel<<<1, 32, 0, stream>>>(ws);
    sl_min_pairdist_kernel<<<64, 256, 0, stream>>>(x, ws);
    sl_finalize_kernel<<<1, 64, 0, stream>>>(Y, ws, out);
}